// KMeans_13460427505707
// MI455X (gfx1250) — compile-verified
//
#include <hip/hip_runtime.h>
#include <stddef.h>

// ---------------- problem constants ----------------
#define BATCH 8
#define NSPK  4
#define DIM   512
#define TLEN  2000
#define KC    4
#define ITERS 100
#define EPSV  1e-8f
#define NT    (NSPK * TLEN)         // 8000 points per batch
#define PTS_PER_WAVE 16
#define WAVES_PER_WG 4
#define PTS_PER_WG   (PTS_PER_WAVE * WAVES_PER_WG)   // 64
#define WGS_PER_B    (NT / PTS_PER_WG)               // 125
#define CROW 514                    // padded LDS row stride for centers (bank-conflict free)

typedef __attribute__((ext_vector_type(2))) float v2f;
typedef __attribute__((ext_vector_type(8))) float v8f;

// ============================================================
// threefry2x32 (faithful to JAX) for kmeans++ init RNG
// ============================================================
__device__ __forceinline__ unsigned rotl32(unsigned x, int r) {
    return (x << r) | (x >> (32 - r));
}

__device__ void threefry2x32(unsigned k0, unsigned k1, unsigned x0, unsigned x1,
                             unsigned* o0, unsigned* o1) {
    unsigned ks0 = k0, ks1 = k1, ks2 = 0x1BD11BDAu ^ k0 ^ k1;
    unsigned ks[3] = {ks0, ks1, ks2};
    const int rot[8] = {13, 15, 26, 6, 17, 29, 16, 24};
    x0 += ks[0]; x1 += ks[1];
#pragma unroll
    for (int r = 0; r < 20; ++r) {
        int ridx = ((r >> 2) & 1) * 4 + (r & 3);
        x0 += x1;
        x1 = rotl32(x1, rot[ridx]);
        x1 ^= x0;
        if ((r & 3) == 3) {
            int j = (r >> 2) + 1;
            x0 += ks[j % 3];
            x1 += ks[(j + 1) % 3] + (unsigned)j;
        }
    }
    *o0 = x0; *o1 = x1;
}

// key,sub = jax.random.split(key): counts [0,1,2,3] -> pairs (0,2),(1,3)
__device__ void jax_split(unsigned k0, unsigned k1,
                          unsigned* nk0, unsigned* nk1, unsigned* s0, unsigned* s1) {
    unsigned y0a, y1a, y0b, y1b;
    threefry2x32(k0, k1, 0u, 2u, &y0a, &y1a);
    threefry2x32(k0, k1, 1u, 3u, &y0b, &y1b);
    *nk0 = y0a; *nk1 = y0b;   // key  = out[0] = [x0-stream word0, word1]
    *s0  = y1a; *s1  = y1b;   // sub  = out[1]
}

// r = jax.random.uniform(sub, (8,)): bits = threefry(sub, iota(8)), pairs (b,b+4)
__device__ float jax_uniform8(unsigned s0, unsigned s1, int b) {
    unsigned lo = (unsigned)(b & 3), hi = lo + 4u;
    unsigned y0, y1;
    threefry2x32(s0, s1, lo, hi, &y0, &y1);
    unsigned bits = (b < 4) ? y0 : y1;
    unsigned f = (bits >> 9) | 0x3f800000u;
    return __uint_as_float(f) - 1.0f;
}

// ============================================================
// kmeans++ initialization: one block per batch
// ============================================================
__global__ __launch_bounds__(256)
void init_kernel(const float* __restrict__ x, float* __restrict__ centers) {
    const int b   = blockIdx.x;
    const int tid = threadIdx.x;

    __shared__ float s_c[KC * DIM];     // 8 KB
    __shared__ float s_csq[KC];
    __shared__ float s_dmin[NT];        // 32 KB
    __shared__ int   s_sel;
    __shared__ float s_r;

    // centers[:,0] = x[b, 0, :, 0]; rows 1..3 zero
    for (int d = tid; d < DIM; d += 256)
        s_c[d] = x[(((size_t)b * NSPK + 0) * DIM + d) * TLEN + 0];
    for (int i = DIM + tid; i < KC * DIM; i += 256) s_c[i] = 0.0f;
    if (tid < KC) s_csq[tid] = 0.0f;
    __syncthreads();
    if (tid == 0) {
        float s = 0.f;
        for (int d = 0; d < DIM; ++d) s += s_c[d] * s_c[d];
        s_csq[0] = s;
    }
    __syncthreads();

    // RNG chain: key(1) = [0,1]
    unsigned k0 = 0u, k1 = 1u;

    for (int i = 1; i < KC; ++i) {
        // per-point dmin against centers 0..i (row i is zero -> dist = fsq)
        for (int p = tid; p < NT; p += 256) {
            int n = p / TLEN, t = p - n * TLEN;
            const float* xp = x + ((size_t)(b * NSPK + n) * DIM) * TLEN + t;
            float fsq = 0.f, dot0 = 0.f, dot1 = 0.f, dot2 = 0.f;
            for (int d = 0; d < DIM; ++d) {
                float v = xp[(size_t)d * TLEN];
                fsq += v * v;
                dot0 += v * s_c[0 * DIM + d];
                dot1 += v * s_c[1 * DIM + d];
                dot2 += v * s_c[2 * DIM + d];
            }
            float dmin = fsq;  // zero-center row included by reference
            float c0 = fsq + s_csq[0] - 2.f * dot0;
            float c1 = fsq + s_csq[1] - 2.f * dot1;
            float c2 = fsq + s_csq[2] - 2.f * dot2;
            dmin = fminf(dmin, c0);
            dmin = fminf(dmin, c1);  // rows >= i are zero -> candidate == fsq (harmless)
            dmin = fminf(dmin, c2);
            s_dmin[p] = dmin;
        }
        __syncthreads();

        if (tid == 0) {
            // advance RNG: key, sub = split(key); r = uniform(sub,(8,))[b]
            unsigned nk0, nk1, s0, s1;
            jax_split(k0, k1, &nk0, &nk1, &s0, &s1);
            k0 = nk0; k1 = nk1;
            float r = jax_uniform8(s0, s1, b);
            float total = 0.f;
            for (int p = 0; p < NT; ++p) total += s_dmin[p];
            float csum = 0.f;
            int sel = -1;
            for (int p = 0; p < NT; ++p) {
                csum += s_dmin[p] / total;
                if (csum <= r) sel++;
            }
            s_sel = sel;
            s_r = r;
        }
        __syncthreads();

        int sel = s_sel;
        int p = sel >= 0 ? sel : 0;
        int n = p / TLEN, t = p - n * TLEN;
        for (int d = tid; d < DIM; d += 256) {
            float v = (sel >= 0)
                ? x[(((size_t)b * NSPK + n) * DIM + d) * TLEN + t] : 0.0f;
            s_c[i * DIM + d] = v;
        }
        __syncthreads();
        if (tid == 0) {
            float s = 0.f;
            for (int d = 0; d < DIM; ++d) s += s_c[i * DIM + d] * s_c[i * DIM + d];
            s_csq[i] = s;
        }
        __syncthreads();
    }

    for (int i = tid; i < KC * DIM; i += 256)
        centers[(size_t)b * KC * DIM + i] = s_c[i];
}

// ============================================================
// prepare: halfsq[b][k] = 0.5 * sum_d c^2
// ============================================================
__global__ __launch_bounds__(128)
void prepare_kernel(const float* __restrict__ centers, float* __restrict__ halfsq) {
    const int bk = blockIdx.x;            // b*K + k
    const int tid = threadIdx.x;
    __shared__ float red[128];
    float s = 0.f;
    for (int d = tid; d < DIM; d += 128) {
        float c = centers[(size_t)bk * DIM + d];
        s += c * c;
    }
    red[tid] = s;
    __syncthreads();
    for (int off = 64; off > 0; off >>= 1) {
        if (tid < off) red[tid] += red[tid + off];
        __syncthreads();
    }
    if (tid == 0) halfsq[bk] = 0.5f * red[0];
}

// ============================================================
// fused assign (WMMA f32 16x16x4) + per-WG partial accumulation
// grid (WGS_PER_B, BATCH), block 128 (4 waves, wave32)
// ============================================================
__global__ __launch_bounds__(128)
void assign_accum_kernel(const float* __restrict__ x,
                         const float* __restrict__ centers,
                         const float* __restrict__ halfsq,
                         float* __restrict__ pSums,
                         float* __restrict__ pCnt) {
    const int b    = blockIdx.y;
    const int wg   = blockIdx.x;
    const int tid  = threadIdx.x;
    const int wave = tid >> 5;
    const int lane = tid & 31;

    __shared__ float s_cneg[16 * CROW];   // negated centers, rows 4..15 zero (~32.9 KB)
    __shared__ float s_hsq[16];
    __shared__ int   s_assign[PTS_PER_WG];
    __shared__ int   s_cnt[KC];

    // stage negated centers (B operand source), zero-pad cols 4..15
    for (int idx = tid; idx < 16 * DIM; idx += 128) {
        int n = idx >> 9, d = idx & (DIM - 1);
        float v = (n < KC) ? -centers[((size_t)b * KC + n) * DIM + d] : 0.0f;
        s_cneg[n * CROW + d] = v;
    }
    if (tid < 16) s_hsq[tid] = (tid < KC) ? halfsq[b * KC + tid] : __builtin_inff();
    if (tid < KC) s_cnt[tid] = 0;
    __syncthreads();

    // ---------- pass 1: distances via WMMA ----------
    const int m     = lane & 15;          // matrix row (point within tile)
    const int kd_hi = lane >> 4;          // 0 for lanes 0-15, 1 for lanes 16-31
    const int P0    = wg * PTS_PER_WG + wave * PTS_PER_WAVE;
    {
        int p = P0 + m;
        int n = p / TLEN, t = p - n * TLEN;
        const float* xp = x + ((size_t)(b * NSPK + n) * DIM) * TLEN + t;
        const float* cb = s_cneg + (lane & 15) * CROW;

        float hc = s_hsq[lane & 15];      // C init: 0.5*sum(c^2) per column, INF for cols>=4
        v8f acc = {hc, hc, hc, hc, hc, hc, hc, hc};

#pragma unroll 2
        for (int d0 = 0; d0 < DIM; d0 += 4) {
            // A (16x4 f32): lane m, K-dim = 2*kd_hi + v
            v2f A;
            A.x = xp[(size_t)(d0 + 2 * kd_hi + 0) * TLEN];
            A.y = xp[(size_t)(d0 + 2 * kd_hi + 1) * TLEN];
            // B (4x16, negated centers): K-dim = 2*v + kd_hi, col = lane&15
            v2f Bv;
            Bv.x = cb[d0 + kd_hi];
            Bv.y = cb[d0 + 2 + kd_hi];
            // D = A*(-C^T) + C0  ->  0.5*sum(c^2) - dot(x, c)
            acc = __builtin_amdgcn_wmma_f32_16x16x4_f32(
                false, A, false, Bv, (short)0, acc, false, false);
        }

        // argmin over columns 0..3 for each of this lane-half's 8 rows.
        // xor-1/xor-2 reduce within groups of 4 lanes; groups {0..3} and {16..19}
        // carry the real columns (lane&15 == column).
#pragma unroll
        for (int j = 0; j < 8; ++j) {
            float v = acc[j];
            int kk = lane & 15;
#pragma unroll
            for (int off = 1; off < 4; off <<= 1) {
                float ov = __shfl_xor(v, off);
                int   ok = __shfl_xor(kk, off);
                if (ov < v || (ov == v && ok < kk)) { v = ov; kk = ok; }
            }
            if (lane == 0)       s_assign[wave * 16 + j]     = kk;  // rows 0..7
            else if (lane == 16) s_assign[wave * 16 + 8 + j] = kk;  // rows 8..15
        }
    }
    __syncthreads();

    // counts
    if (tid < PTS_PER_WG) atomicAdd(&s_cnt[s_assign[tid]], 1);
    __syncthreads();
    if (tid < KC)
        pCnt[((size_t)b * WGS_PER_B + wg) * KC + tid] = (float)s_cnt[tid];

    // ---------- pass 2: per-cluster partial sums (L0/L2-hot re-read) ----------
    // thread owns d in {tid, tid+128, tid+256, tid+384}; 16 named accumulators
    float a00=0,a01=0,a02=0,a03=0;
    float a10=0,a11=0,a12=0,a13=0;
    float a20=0,a21=0,a22=0,a23=0;
    float a30=0,a31=0,a32=0,a33=0;
    const int pb = wg * PTS_PER_WG;
#pragma unroll 1
    for (int p = 0; p < PTS_PER_WG; ++p) {
        int k = s_assign[p];              // uniform across the block
        int P = pb + p;
        int n = P / TLEN, t = P - n * TLEN;
        const float* xp = x + ((size_t)(b * NSPK + n) * DIM) * TLEN + t;
        float v0 = xp[(size_t)(tid      ) * TLEN];
        float v1 = xp[(size_t)(tid + 128) * TLEN];
        float v2 = xp[(size_t)(tid + 256) * TLEN];
        float v3 = xp[(size_t)(tid + 384) * TLEN];
        if      (k == 0) { a00+=v0; a01+=v1; a02+=v2; a03+=v3; }
        else if (k == 1) { a10+=v0; a11+=v1; a12+=v2; a13+=v3; }
        else if (k == 2) { a20+=v0; a21+=v1; a22+=v2; a23+=v3; }
        else             { a30+=v0; a31+=v1; a32+=v2; a33+=v3; }
    }
    float* dst = pSums + ((size_t)(b * WGS_PER_B + wg) * KC) * DIM;
    dst[0*DIM + tid      ] = a00; dst[0*DIM + tid + 128] = a01;
    dst[0*DIM + tid + 256] = a02; dst[0*DIM + tid + 384] = a03;
    dst[1*DIM + tid      ] = a10; dst[1*DIM + tid + 128] = a11;
    dst[1*DIM + tid + 256] = a12; dst[1*DIM + tid + 384] = a13;
    dst[2*DIM + tid      ] = a20; dst[2*DIM + tid + 128] = a21;
    dst[2*DIM + tid + 256] = a22; dst[2*DIM + tid + 384] = a23;
    dst[3*DIM + tid      ] = a30; dst[3*DIM + tid + 128] = a31;
    dst[3*DIM + tid + 256] = a32; dst[3*DIM + tid + 384] = a33;
}

// ============================================================
// update: deterministic reduction over 125 WG partials
// ============================================================
__global__ __launch_bounds__(256)
void update_kernel(const float* __restrict__ pSums,
                   const float* __restrict__ pCnt,
                   float* __restrict__ out) {
    const int g = blockIdx.x * 256 + threadIdx.x;  // b*K*D + k*D + d
    const int b = g / (KC * DIM);
    const int k = (g / DIM) % KC;
    const int d = g % DIM;
    float s = 0.f, c = 0.f;
#pragma unroll 1
    for (int wg = 0; wg < WGS_PER_B; ++wg) {
        s += pSums[(((size_t)b * WGS_PER_B + wg) * KC + k) * DIM + d];
        c += pCnt[((size_t)b * WGS_PER_B + wg) * KC + k];
    }
    out[g] = s / (c + EPSV);
}

// ============================================================
// host launcher
// ============================================================
extern "C" void kernel_launch(void* const* d_in, const int* in_sizes, int n_in,
                              void* d_out, int out_size, void* d_ws, size_t ws_size,
                              hipStream_t stream) {
    (void)in_sizes; (void)n_in; (void)out_size; (void)ws_size;
    const float* x = (const float*)d_in[0];
    float* out = (float*)d_out;

    // workspace layout (floats): centers | halfsq | pCnt | pSums  (~8.3 MB)
    float* centers = (float*)d_ws;
    float* halfsq  = centers + (size_t)BATCH * KC * DIM;                 // 16384
    float* pCnt    = halfsq + BATCH * KC;                                // +32
    float* pSums   = pCnt + (size_t)BATCH * WGS_PER_B * KC;              // +4000

    init_kernel<<<BATCH, 256, 0, stream>>>(x, centers);

    for (int it = 0; it < ITERS; ++it) {
        prepare_kernel<<<BATCH * KC, 128, 0, stream>>>(centers, halfsq);
        assign_accum_kernel<<<dim3(WGS_PER_B, BATCH), 128, 0, stream>>>(
            x, centers, halfsq, pSums, pCnt);
        update_kernel<<<(BATCH * KC * DIM) / 256, 256, 0, stream>>>(
            pSums, pCnt, (it == ITERS - 1) ? out : centers);
    }
}